// DGLSAGE_67130338837023
// MI455X (gfx1250) — compile-verified
//
#include <hip/hip_runtime.h>
#include <hip/hip_bf16.h>

#define GNN_N_NODES 50000
#define GNN_DIM_IN  96
#define GNN_DIM_HID 96
#define GNN_DIM_OUT 64

typedef __attribute__((ext_vector_type(2))) float v2f;
typedef __attribute__((ext_vector_type(8))) float v8f;

// -------------------------------------------------------------------------
// Degree: deg[dst[e]] += 1  (depends only on dst; computed once)
// -------------------------------------------------------------------------
__global__ void sage_degree(const int* __restrict__ dst, float* __restrict__ deg,
                            int n_edges) {
    int i = blockIdx.x * blockDim.x + threadIdx.x;
    if (i < n_edges) atomicAdd(&deg[dst[i]], 1.0f);
}

// -------------------------------------------------------------------------
// Segment sum: msum[dst[e], :] += h[src[e], :]
// blockDim.x == feature dim (96 = 3 waves) -> coalesced row gather and
// consecutive-address f32 atomics (global_atomic_add_f32), L2-resident.
// -------------------------------------------------------------------------
__global__ void sage_scatter(const float* __restrict__ H,
                             const int* __restrict__ src,
                             const int* __restrict__ dst,
                             float* __restrict__ msum,
                             int n_edges, int d) {
    for (int e = blockIdx.x; e < n_edges; e += gridDim.x) {
        const int s = src[e];
        const int t = dst[e];
        const float v = H[(size_t)s * d + threadIdx.x];
        atomicAdd(&msum[(size_t)t * d + threadIdx.x], v);
    }
}

// -------------------------------------------------------------------------
// Fused SAGE layer GEMM using V_WMMA_F32_16X16X4_F32 (fp32 matrix cores):
//   OUT = act( H @ Wself^T + (msum * 1/max(deg,1)) @ Wneigh^T + bias )
// One wave computes a 16-node x 16-outfeature tile; 8 waves per block share
// an LDS-staged weight tile for one out-feature tile.
//
// WMMA operand maps (wave32, per CDNA5 ISA 7.12.2):
//   A (16x4 f32, v2f): lanes 0-15 -> M=lane, {.x=K+0,.y=K+1};
//                      lanes 16-31 -> M=lane-16, {.x=K+2,.y=K+3}
//   B (4x16 f32, v2f): mirror of A with N instead of M
//   C/D (16x16 f32, v8f): vgpr r, lane L -> M = r + 8*(L>=16), N = L%16
// -------------------------------------------------------------------------
template <int DIN, int DOUT, bool RELU>
__global__ __launch_bounds__(256)
void sage_gemm(const float* __restrict__ H,       // [N, DIN]
               const float* __restrict__ MSUM,    // [N, DIN]
               const float* __restrict__ DEG,     // [N]
               const float* __restrict__ Wself,   // [DOUT, DIN]
               const float* __restrict__ Wneigh,  // [DOUT, DIN]
               const float* __restrict__ Bias,    // [DOUT]
               float* __restrict__ OUT,           // [N, DOUT]
               int n_nodes) {
    __shared__ float ws_tile[DIN * 16];   // [k][n] layout
    __shared__ float wn_tile[DIN * 16];
    __shared__ float b_tile[16];

    const int lane  = threadIdx.x & 31;
    const int wave  = threadIdx.x >> 5;                    // 0..7
    const int otile = blockIdx.y;                          // out-feature tile
    const int nbase = (blockIdx.x * 8 + wave) * 16;        // node tile base

    // Stage weight tiles for this out-feature tile into LDS
    for (int i = threadIdx.x; i < DIN * 16; i += blockDim.x) {
        const int k = i >> 4;
        const int n = i & 15;
        ws_tile[i] = Wself[(size_t)(otile * 16 + n) * DIN + k];
        wn_tile[i] = Wneigh[(size_t)(otile * 16 + n) * DIN + k];
    }
    if (threadIdx.x < 16) b_tile[threadIdx.x] = Bias[otile * 16 + threadIdx.x];
    __syncthreads();

    if (nbase >= n_nodes) return;   // whole-wave exit: EXEC stays all-1s

    const int half = lane >> 4;      // 0 -> K+0/+1, 1 -> K+2/+3
    const int mrow = lane & 15;      // M row (A) / N col (B,C,D)
    const int row  = nbase + mrow;

    const float rdeg = 1.0f / fmaxf(DEG[row], 1.0f);
    const float* __restrict__ hrow = H    + (size_t)row * DIN;
    const float* __restrict__ srow = MSUM + (size_t)row * DIN;

    // Seed accumulator with bias (same value in all 8 row-VGPRs of a lane)
    v8f c;
    const float bv = b_tile[mrow];
    #pragma unroll
    for (int r = 0; r < 8; ++r) c[r] = bv;

    #pragma unroll
    for (int k = 0; k < DIN; k += 4) {
        const int ka = k + 2 * half;
        v2f a_s, a_n, b_s, b_n;
        a_s.x = hrow[ka];
        a_s.y = hrow[ka + 1];
        a_n.x = srow[ka] * rdeg;           // fold mean-normalization into A
        a_n.y = srow[ka + 1] * rdeg;
        b_s.x = ws_tile[ka * 16 + mrow];
        b_s.y = ws_tile[(ka + 1) * 16 + mrow];
        b_n.x = wn_tile[ka * 16 + mrow];
        b_n.y = wn_tile[(ka + 1) * 16 + mrow];
        c = __builtin_amdgcn_wmma_f32_16x16x4_f32(false, a_s, false, b_s,
                                                  (short)0, c, false, false);
        c = __builtin_amdgcn_wmma_f32_16x16x4_f32(false, a_n, false, b_n,
                                                  (short)0, c, false, false);
    }

    const int ncol   = otile * 16 + mrow;
    const int rowoff = nbase + 8 * half;
    #pragma unroll
    for (int r = 0; r < 8; ++r) {
        float v = c[r];
        if (RELU) v = fmaxf(v, 0.0f);
        OUT[(size_t)(rowoff + r) * DOUT + ncol] = v;
    }
}

// -------------------------------------------------------------------------
// Host orchestration (graph-capture safe: only async ops on `stream`)
// -------------------------------------------------------------------------
extern "C" void kernel_launch(void* const* d_in, const int* in_sizes, int n_in,
                              void* d_out, int out_size, void* d_ws, size_t ws_size,
                              hipStream_t stream) {
    const float* x   = (const float*)d_in[0];
    const int*   src = (const int*)d_in[1];
    const int*   dst = (const int*)d_in[2];
    const float* ws0 = (const float*)d_in[3];
    const float* wn0 = (const float*)d_in[4];
    const float* b0  = (const float*)d_in[5];
    const float* ws1 = (const float*)d_in[6];
    const float* wn1 = (const float*)d_in[7];
    const float* b1  = (const float*)d_in[8];
    const float* ws2 = (const float*)d_in[9];
    const float* wn2 = (const float*)d_in[10];
    const float* b2  = (const float*)d_in[11];

    const int n_nodes = in_sizes[0] / GNN_DIM_IN;   // 50000
    const int n_edges = in_sizes[1];                // 800000

    // Workspace carve-out
    float* msum = (float*)d_ws;                                  // [N, 96]
    float* deg  = msum + (size_t)n_nodes * GNN_DIM_HID;          // [N]
    float* h1   = deg + n_nodes;                                 // [N, 96]
    float* h2   = h1 + (size_t)n_nodes * GNN_DIM_HID;            // [N, 96]

    const size_t msum_bytes = (size_t)n_nodes * GNN_DIM_HID * sizeof(float);

    // Degrees depend only on dst: compute once.
    hipMemsetAsync(deg, 0, (size_t)n_nodes * sizeof(float), stream);
    sage_degree<<<(n_edges + 255) / 256, 256, 0, stream>>>(dst, deg, n_edges);

    const int ntiles = (n_nodes + 15) / 16;
    const dim3 gemm_block(256);
    const dim3 grid_hid((ntiles + 7) / 8, GNN_DIM_HID / 16);
    const dim3 grid_out((ntiles + 7) / 8, GNN_DIM_OUT / 16);
    const int scat_blocks = 8192;

    // ---- Layer 0: h1 = relu(sage(x)) ----
    hipMemsetAsync(msum, 0, msum_bytes, stream);
    sage_scatter<<<scat_blocks, GNN_DIM_IN, 0, stream>>>(x, src, dst, msum,
                                                         n_edges, GNN_DIM_IN);
    sage_gemm<GNN_DIM_IN, GNN_DIM_HID, true>
        <<<grid_hid, gemm_block, 0, stream>>>(x, msum, deg, ws0, wn0, b0, h1, n_nodes);

    // ---- Layer 1: h2 = relu(sage(h1)) ----
    hipMemsetAsync(msum, 0, msum_bytes, stream);
    sage_scatter<<<scat_blocks, GNN_DIM_HID, 0, stream>>>(h1, src, dst, msum,
                                                          n_edges, GNN_DIM_HID);
    sage_gemm<GNN_DIM_HID, GNN_DIM_HID, true>
        <<<grid_hid, gemm_block, 0, stream>>>(h1, msum, deg, ws1, wn1, b1, h2, n_nodes);

    // ---- Layer 2: out = sage(h2), no activation ----
    hipMemsetAsync(msum, 0, msum_bytes, stream);
    sage_scatter<<<scat_blocks, GNN_DIM_HID, 0, stream>>>(h2, src, dst, msum,
                                                          n_edges, GNN_DIM_HID);
    sage_gemm<GNN_DIM_HID, GNN_DIM_OUT, false>
        <<<grid_out, gemm_block, 0, stream>>>(h2, msum, deg, ws2, wn2, b2,
                                              (float*)d_out, n_nodes);
}